// NPMLPHead_53893249630457
// MI455X (gfx1250) — compile-verified
//
#include <hip/hip_runtime.h>
#include <cstddef>
#include <cstdint>

typedef __attribute__((ext_vector_type(2))) float v2f;
typedef __attribute__((ext_vector_type(8))) float v8f;

#define NPATCH 128
#define NCH    256
#define NBATCH 16

// K-chunk for staging the gathered X tile in LDS (same for all levels)
#define KC  256
#define XS  257   // padded LDS stride for X/Y tiles (257 mod 64 == 1 -> low conflicts)
#define HS  257   // padded LDS stride for H tile

struct NPArgs {
  const float* feat[3];
  const int*   idx[3];
  const float* w1[3];
  const float* b1[3];
  const float* w2[3];
  const float* b2[3];
  float*       out;   // [3, 16, 128, 256]
};

// Wait for all outstanding async (global<->LDS) transfers of this wave.
__device__ __forceinline__ void wait_async0() {
#if __has_builtin(__builtin_amdgcn_s_wait_asynccnt)
  __builtin_amdgcn_s_wait_asynccnt(0);
#else
  asm volatile("s_wait_asynccnt 0x0" ::: "memory");
#endif
}

// Issue one X chunk [16 rows x KC channels] as async scattered loads straight
// into LDS (GLOBAL_LOAD_ASYNC_TO_LDS_B32, tracked with ASYNCcnt). Each of the
// 512 threads owns row j = tid&15, channels c0+32k (c0 = tid>>4, k = 0..7):
// 8 cachelines in flight per lane with zero VGPR data movement.
template<int HW2>
__device__ __forceinline__
void issue_gather_async(const float* __restrict__ fkc, int plane_off,
                        float* buf, int tid)
{
  const int j  = tid & 15;
  const int c0 = tid >> 4;                       // 0..31
  const float* g = fkc + (size_t)c0 * HW2 + plane_off;
  // Low 32 bits of a flat shared pointer == LDS byte address (aperture is in
  // bits 63:32), which is exactly the VDST operand of async-to-LDS.
  const unsigned lds0 = (unsigned)(uintptr_t)(buf + j * XS + c0);
  #pragma unroll
  for (int k = 0; k < 8; ++k) {
    asm volatile("global_load_async_to_lds_b32 %0, %1, off"
                 :: "v"(lds0 + (unsigned)(k * 32 * 4)),
                    "v"(g + (size_t)k * 32 * HW2)
                 : "memory");
  }
}

template<int C, int HW>
__device__ __forceinline__
void level_body(const float* __restrict__ feat, const int* __restrict__ idx,
                const float* __restrict__ w1,   const float* __restrict__ b1,
                const float* __restrict__ w2,   const float* __restrict__ b2,
                float* __restrict__ out,
                float* s_x0, float* s_x1, float* s_h, int* s_poff)
{
  constexpr int HW2 = HW * HW;

  const int tid   = threadIdx.x;
  const int lane  = tid & 31;
  const int wv    = tid >> 5;            // wave id 0..15 -> N tile
  const int r0    = blockIdx.x * 16;     // global row tile (rows = b*128 + p)
  const int b     = r0 >> 7;             // 128 patches per batch; tiles never span b
  const int p0    = r0 & 127;

  // WMMA fp32 16x16x4 lane mapping (ISA 7.12.2):
  //   A (16x4): lane -> M = lane&15 ; VGPR v + lane-half -> K = (lane>>4)*2 + v
  //   C/D     : VGPR v -> row v + (lane>>4)*8 ; col = lane&15
  const int mrow  = lane & 15;
  const int khalf = (lane >> 4) << 1;    // 0 or 2
  const int n0    = wv << 4;

  if (tid < 16) s_poff[tid] = idx[p0 + tid];
  __syncthreads();
  const int plane_off = s_poff[tid & 15];        // this thread's gather row offset

  const float* fb = feat + (size_t)b * C * HW2;
  float* bufs[2] = { s_x0, s_x1 };

  // Prologue: kick off the DMA for chunk 0.
  issue_gather_async<HW2>(fb, plane_off, bufs[0], tid);

  // ---------------- GEMM1: H = relu(X @ w1 + b1), K = C ----------------
  v8f acc;
  {
    const float bias1 = b1[n0 + mrow];
    #pragma unroll
    for (int v = 0; v < 8; ++v) acc[v] = bias1;
  }

  int cur = 0;
  for (int kc = 0; kc < C; kc += KC) {
    wait_async0();        // this wave's chunk-kc loads landed in LDS
    __syncthreads();      // everyone's landed; prior reads of the other buf done

    // Overlap: start the DMA of the next chunk into the other buffer while we
    // run WMMA on the current one.
    if (kc + KC < C)
      issue_gather_async<HW2>(fb + (size_t)(kc + KC) * HW2, plane_off,
                              bufs[cur ^ 1], tid);

    const float* xb  = bufs[cur];
    const float* w1k = w1 + (size_t)kc * NCH;
    #pragma unroll 4
    for (int k0 = 0; k0 < KC; k0 += 4) {
      v2f afrag, bfrag;
      afrag[0] = xb[mrow * XS + k0 + khalf + 0];
      afrag[1] = xb[mrow * XS + k0 + khalf + 1];
      bfrag[0] = w1k[(size_t)(k0 + khalf + 0) * NCH + n0 + mrow];
      bfrag[1] = w1k[(size_t)(k0 + khalf + 1) * NCH + n0 + mrow];
      acc = __builtin_amdgcn_wmma_f32_16x16x4_f32(
          /*neg_a=*/false, afrag, /*neg_b=*/false, bfrag,
          /*c_mod=*/(short)0, acc, /*reuse_a=*/false, /*reuse_b=*/false);
    }
    cur ^= 1;
  }

  // ReLU + stage H tile [16 x 256] in LDS
  #pragma unroll
  for (int v = 0; v < 8; ++v) {
    float h = acc[v];
    h = h > 0.0f ? h : 0.0f;
    s_h[(v + ((lane >> 4) << 3)) * HS + n0 + mrow] = h;
  }
  __syncthreads();   // all waves done with GEMM1 (and all s_x reads / asyncs)

  // ---------------- GEMM2: Y = H @ w2 + b2, K = 256 ----------------
  v8f acc2;
  {
    const float bias2 = b2[n0 + mrow];
    #pragma unroll
    for (int v = 0; v < 8; ++v) acc2[v] = bias2;
  }
  #pragma unroll 4
  for (int k0 = 0; k0 < NCH; k0 += 4) {
    v2f afrag, bfrag;
    afrag[0] = s_h[mrow * HS + k0 + khalf + 0];
    afrag[1] = s_h[mrow * HS + k0 + khalf + 1];
    bfrag[0] = w2[(size_t)(k0 + khalf + 0) * NCH + n0 + mrow];
    bfrag[1] = w2[(size_t)(k0 + khalf + 1) * NCH + n0 + mrow];
    acc2 = __builtin_amdgcn_wmma_f32_16x16x4_f32(
        false, afrag, false, bfrag, (short)0, acc2, false, false);
  }

  // stage Y tile in LDS (aliases freed X buffer 0: everyone passed the H barrier)
  float* s_y = s_x0;
  #pragma unroll
  for (int v = 0; v < 8; ++v)
    s_y[(v + ((lane >> 4) << 3)) * HS + n0 + mrow] = acc2[v];
  __syncthreads();

  // ---------------- L2 normalize: wave wv owns row wv ----------------
  const float* yr = s_y + wv * HS;
  float s = 0.0f;
  #pragma unroll
  for (int k = 0; k < 8; ++k) { const float t = yr[lane + (k << 5)]; s += t * t; }
  #pragma unroll
  for (int off = 16; off > 0; off >>= 1) s += __shfl_xor(s, off, 32);
  const float scale = 1.0f / (sqrtf(s) + 1e-7f);

  float* orow = out + ((size_t)b * NPATCH + (p0 + wv)) * NCH;
  #pragma unroll
  for (int k = 0; k < 8; ++k) {
    const int n = lane + (k << 5);
    orow[n] = yr[n] * scale;
  }
}

__global__ __launch_bounds__(512)
void np_mlp_head_kernel(NPArgs a)
{
  __shared__ float s_x0[16 * XS];   // X ping  (later reused as Y tile)
  __shared__ float s_x1[16 * XS];   // X pong
  __shared__ float s_h [16 * HS];
  __shared__ int   s_poff[16];

  const size_t lvl_stride = (size_t)NBATCH * NPATCH * NCH;
  switch (blockIdx.y) {
    case 0:
      level_body<256, 128>(a.feat[0], a.idx[0], a.w1[0], a.b1[0], a.w2[0], a.b2[0],
                           a.out + 0 * lvl_stride, s_x0, s_x1, s_h, s_poff);
      break;
    case 1:
      level_body<512, 64>(a.feat[1], a.idx[1], a.w1[1], a.b1[1], a.w2[1], a.b2[1],
                          a.out + 1 * lvl_stride, s_x0, s_x1, s_h, s_poff);
      break;
    default:
      level_body<1024, 32>(a.feat[2], a.idx[2], a.w1[2], a.b1[2], a.w2[2], a.b2[2],
                           a.out + 2 * lvl_stride, s_x0, s_x1, s_h, s_poff);
      break;
  }
}

extern "C" void kernel_launch(void* const* d_in, const int* in_sizes, int n_in,
                              void* d_out, int out_size, void* d_ws, size_t ws_size,
                              hipStream_t stream) {
  (void)in_sizes; (void)n_in; (void)out_size; (void)d_ws; (void)ws_size;

  NPArgs a;
  // setup_inputs() dict order: per level i -> feat, idx, w1, b1, w2, b2
  for (int i = 0; i < 3; ++i) {
    a.feat[i] = (const float*)d_in[6 * i + 0];
    a.idx[i]  = (const int*)  d_in[6 * i + 1];
    a.w1[i]   = (const float*)d_in[6 * i + 2];
    a.b1[i]   = (const float*)d_in[6 * i + 3];
    a.w2[i]   = (const float*)d_in[6 * i + 4];
    a.b2[i]   = (const float*)d_in[6 * i + 5];
  }
  a.out = (float*)d_out;

  dim3 grid(128, 3, 1);   // 128 row-tiles of 16 rows x 3 levels
  dim3 block(512, 1, 1);  // 16 waves (wave32): one wave per 16-col tile
  np_mlp_head_kernel<<<grid, block, 0, stream>>>(a);
}